// ChamferPccRateDistortionLoss_74560632259515
// MI455X (gfx1250) — compile-verified
//
#include <hip/hip_runtime.h>
#include <hip/hip_bf16.h>
#include <float.h>
#include <math.h>

typedef float v2f __attribute__((ext_vector_type(2)));
typedef float v8f __attribute__((ext_vector_type(8)));

#define NBATCH 8

// Both directions of Chamfer in one launch: first half of the grid computes
// min_q ||x_p - y_q||^2 -> minx, second half computes min_p ||y_q - x_p||^2
// -> miny. Each wave owns a 16-row tile of X; loops over all 16-col tiles
// of Y. Dot products via V_WMMA_F32_16X16X4_F32 (K: 3 coords + zero pad).
__global__ __launch_bounds__(256)
void chamfer_min_kernel(const float* __restrict__ x_hat,
                        const float* __restrict__ points,
                        float* __restrict__ minx,
                        float* __restrict__ miny, int P) {
    const int halfGrid = gridDim.x >> 1;
    const int dir = (blockIdx.x >= halfGrid) ? 1 : 0;   // uniform per block
    const int bid = blockIdx.x - dir * halfGrid;

    const float* __restrict__ X = dir ? points : x_hat;
    const float* __restrict__ Y = dir ? x_hat  : points;
    float* __restrict__ rowmin  = dir ? miny   : minx;

    const int wave = threadIdx.x >> 5;
    const int lane = threadIdx.x & 31;
    const int half = lane >> 4;        // 0: lanes 0-15 (K=0,1), 1: lanes 16-31 (K=2,3)
    const int l16  = lane & 15;
    const int tilesPerBatch = P / 128; // 8 waves * 16 rows
    const int n  = bid / tilesPerBatch;
    const int p0 = (bid % tilesPerBatch) * 128 + wave * 16;

    const float* __restrict__ xb = X + (size_t)n * P * 3;
    const float* __restrict__ yb = Y + (size_t)n * P * 3;

    // ---- A fragment: rows p0..p0+15, K = {c0, c1, c2, 0} (16x4 f32 layout) ----
    const int arow = p0 + l16;
    const float ax = xb[arow * 3 + 0];
    const float ay = xb[arow * 3 + 1];
    const float az = xb[arow * 3 + 2];
    v2f a;
    a.x = half ? az  : ax;
    a.y = half ? 0.f : ay;
    const float x2all = ax * ax + ay * ay + az * az;  // |x_row|^2 for row = l16

    // Broadcast |x|^2 for the 8 rows this lane's D registers correspond to:
    // VGPR r of D holds M = r (lanes 0-15) or M = r+8 (lanes 16-31).
    float x2row[8];
#pragma unroll
    for (int r = 0; r < 8; ++r)
        x2row[r] = __shfl(x2all, r + (half ? 8 : 0), 32);

    float rm[8];
#pragma unroll
    for (int r = 0; r < 8; ++r) rm[r] = FLT_MAX;

    for (int q0 = 0; q0 < P; q0 += 16) {
        const int qcol = q0 + l16;   // lane l and l+16 load the same column point
        const float y0 = yb[qcol * 3 + 0];
        const float y1 = yb[qcol * 3 + 1];
        const float yz = yb[qcol * 3 + 2];
        const float y2s = y0 * y0 + y1 * y1 + yz * yz;   // |y_col|^2, per N=l16

        // ---- B fragment: 4x16 (K x N), mirror of A layout ----
        v2f b;
        b.x = half ? yz  : y0;
        b.y = half ? 0.f : y1;

        v8f c = {};
        // D = A x B : 16x16 tile of x . y
        v8f d = __builtin_amdgcn_wmma_f32_16x16x4_f32(
            /*neg_a=*/false, a, /*neg_b=*/false, b,
            /*c_mod=*/(short)0, c, /*reuse_a=*/false, /*reuse_b=*/false);

#pragma unroll
        for (int r = 0; r < 8; ++r) {
            const float dist = x2row[r] + y2s - 2.0f * d[r];
            rm[r] = fminf(rm[r], dist);
        }
    }

    // Min across the 16 lanes of each half (the N dimension of each row).
#pragma unroll
    for (int r = 0; r < 8; ++r) {
        float v = rm[r];
        v = fminf(v, __shfl_xor(v, 1, 32));
        v = fminf(v, __shfl_xor(v, 2, 32));
        v = fminf(v, __shfl_xor(v, 4, 32));
        v = fminf(v, __shfl_xor(v, 8, 32));
        rm[r] = v;
    }
    if (l16 == 0) {
#pragma unroll
        for (int r = 0; r < 8; ++r)
            rowmin[(size_t)n * P + p0 + half * 8 + r] = rm[r];
    }
}

// Single-block reduction + scalar epilogue.
// out = {loss, bpp_loss, rec_loss, bit_loss, bpp_y}
__global__ __launch_bounds__(256)
void finalize_kernel(const float* __restrict__ lik, int NM,
                     const float* __restrict__ minx,
                     const float* __restrict__ miny, int NP,
                     float* __restrict__ out, int P) {
    const int t = threadIdx.x;
    float sL = 0.f, sX = 0.f, sY = 0.f;
    for (int i = t; i < NM; i += 256) sL += log2f(lik[i]);
    for (int i = t; i < NP; i += 256) { sX += minx[i]; sY += miny[i]; }

    // wave32 butterfly reduce all three sums
#pragma unroll
    for (int m = 16; m >= 1; m >>= 1) {
        sL += __shfl_xor(sL, m, 32);
        sX += __shfl_xor(sX, m, 32);
        sY += __shfl_xor(sY, m, 32);
    }
    __shared__ float wL[8], wX[8], wY[8];
    if ((t & 31) == 0) { wL[t >> 5] = sL; wX[t >> 5] = sX; wY[t >> 5] = sY; }
    __syncthreads();
    if (t == 0) {
        float L = 0.f, Xs = 0.f, Ys = 0.f;
#pragma unroll
        for (int w = 0; w < 8; ++w) { L += wL[w]; Xs += wX[w]; Ys += wY[w]; }
        const float bit_y = L / (-(float)NBATCH);
        const float bpp   = bit_y / (float)P;
        const float rec   = Xs / (float)NP + Ys / (float)NP;
        out[0] = bpp + rec;   // loss   (LMBDA = 1.0)
        out[1] = bpp;         // bpp_loss
        out[2] = rec;         // rec_loss
        out[3] = bit_y;       // bit_loss
        out[4] = bpp;         // bpp_y (== bpp_loss analytically)
    }
}

extern "C" void kernel_launch(void* const* d_in, const int* in_sizes, int n_in,
                              void* d_out, int out_size, void* d_ws, size_t ws_size,
                              hipStream_t stream) {
    const float* x_hat  = (const float*)d_in[0];   // [N, P, 3]
    const float* points = (const float*)d_in[1];   // [N, P, 3]
    const float* lik_y  = (const float*)d_in[2];   // [N, M]

    const int P  = in_sizes[0] / (NBATCH * 3);     // 4096
    const int NM = in_sizes[2];                    // N*M
    const int NP = NBATCH * P;

    float* minx = (float*)d_ws;      // [N*P] min_q ||x_p - y_q||^2
    float* miny = minx + NP;         // [N*P] min_p ||y_q - x_p||^2

    const int blocksPerDir = NBATCH * (P / 128);   // 128 rows per workgroup
    // Both chamfer directions in one launch: grid = 2 * blocksPerDir.
    chamfer_min_kernel<<<2 * blocksPerDir, 256, 0, stream>>>(
        x_hat, points, minx, miny, P);
    finalize_kernel<<<1, 256, 0, stream>>>(lik_y, NM, minx, miny, NP,
                                           (float*)d_out, P);
}